// LDAHead_7859790152164
// MI455X (gfx1250) — compile-verified
//
#include <hip/hip_runtime.h>
#include <math.h>

// Problem constants (match reference)
#define B_N 4096
#define C_N 64
#define D_N 256
#define EPSF 1e-5f
#define NTOT (4096.0f + 64.0f * 1e-5f)   // sum of (n_c + eps)

typedef __attribute__((ext_vector_type(2))) float v2f;
typedef __attribute__((ext_vector_type(8))) float v8f;

// Exact-f32 WMMA: D(16x16,f32) = A(16x4,f32) x B(4x16,f32) + C
__device__ __forceinline__ v8f wmma4(v2f a, v2f b, v8f c) {
  return __builtin_amdgcn_wmma_f32_16x16x4_f32(
      /*neg_a=*/false, a, /*neg_b=*/false, b,
      /*c_mod=*/(short)0, c, /*reuse_a=*/false, /*reuse_b=*/false);
}

// ---------------------------------------------------------------------------
// Kernel 1: per-class counts, mean, mean-transpose, log prior, cov weights.
// grid = C_N blocks, 256 threads (thread = feature dim d). Fixed-order sums.
// ---------------------------------------------------------------------------
__global__ void k_class_stats(const float* __restrict__ z, const int* __restrict__ y,
                              float* __restrict__ mu, float* __restrict__ muT,
                              float* __restrict__ wc, float* __restrict__ logprior) {
  const int c = blockIdx.x;
  const int t = threadIdx.x;            // feature dim
  float acc = 0.0f;
  int cnt = 0;
  for (int b = 0; b < B_N; ++b) {
    const int yb = y[b];                // uniform -> scalar branch
    if (yb == c) { acc += z[b * D_N + t]; ++cnt; }
  }
  const float cf = (float)cnt + EPSF;   // counts_c = n_c + eps
  const float m  = acc / cf;
  mu[c * D_N + t]  = m;
  muT[t * C_N + c] = m;
  if (t == 0) {
    wc[c] = cf + EPSF;                  // weight in cov correction = n_c + 2eps
    logprior[c] = logf(cf / NTOT);
  }
}

// ---------------------------------------------------------------------------
// Kernel 2: Gram matrix G = Z^T Z (256x256, K=4096) with f32 WMMA.
// One wave per 16x16 output tile; 4 accumulators to break the WMMA chain.
// grid = 32 blocks x 256 threads (8 waves/block) -> 256 tiles.
// ---------------------------------------------------------------------------
__global__ void k_gram(const float* __restrict__ z, float* __restrict__ G) {
  const int wave = blockIdx.x * (blockDim.x >> 5) + (threadIdx.x >> 5);
  const int lane = threadIdx.x & 31;
  const int m    = lane & 15;
  const int half = lane >> 4;           // 0: K=0,1  1: K=2,3
  const int d0 = (wave >> 4) << 4;      // output row block
  const int e0 = (wave & 15) << 4;      // output col block

  v8f acc[4] = {};
  for (int k0 = 0; k0 < B_N; k0 += 16) {
#pragma unroll
    for (int u = 0; u < 4; ++u) {
      const int kk = k0 + u * 4 + half * 2;
      v2f a, b;
      a.x = z[(kk    ) * D_N + d0 + m];   // A[m][k] = Z^T[d0+m][k] = Z[k][d0+m]
      a.y = z[(kk + 1) * D_N + d0 + m];
      b.x = z[(kk    ) * D_N + e0 + m];   // B[k][n] = Z[k][e0+n]
      b.y = z[(kk + 1) * D_N + e0 + m];
      acc[u] = wmma4(a, b, acc[u]);
    }
  }
  v8f t = acc[0] + acc[1] + acc[2] + acc[3];
#pragma unroll
  for (int v = 0; v < 8; ++v)
    G[(d0 + v + 8 * half) * D_N + e0 + m] = t[v];
}

// ---------------------------------------------------------------------------
// Kernel 3: cov[d][e] = (0.5*(G[d,e]+G[e,d]) - sum_c wc[c]*mu[c,d]*mu[c,e])/N
//                       + eps * (d==e).   grid = 256 (d), block = 256 (e).
// ---------------------------------------------------------------------------
__global__ void k_cov(const float* __restrict__ G, const float* __restrict__ mu,
                      const float* __restrict__ wc, float* __restrict__ covP) {
  const int d = blockIdx.x;
  const int e = threadIdx.x;
  float s = 0.0f;
  for (int c = 0; c < C_N; ++c)
    s = fmaf(wc[c] * mu[c * D_N + d], mu[c * D_N + e], s);
  const float g = 0.5f * (G[d * D_N + e] + G[e * D_N + d]);
  float v = (g - s) / NTOT;
  if (d == e) v += EPSF;
  covP[d * D_N + e] = v;
}

// ---------------------------------------------------------------------------
// Kernel 4: in-place Gauss-Jordan inverse of the SPD 256x256 matrix (-> P).
// Single workgroup, 1024 threads. thread t: column c = t&255, row-group g.
// ---------------------------------------------------------------------------
__global__ void k_gj_inverse(float* __restrict__ A) {
  __shared__ float prow[D_N];
  __shared__ float colv[D_N];
  __shared__ float dd_s;
  const int t = threadIdx.x;
  const int c = t & 255;
  const int g = t >> 8;                 // 4 groups of 64 rows
  for (int j = 0; j < D_N; ++j) {
    if (t == 0) dd_s = 1.0f / A[j * D_N + j];
    __syncthreads();
    const float dd = dd_s;
    if (t < D_N) {
      prow[t] = A[j * D_N + t] * dd;    // scaled pivot row
      colv[t] = A[t * D_N + j];         // old pivot column
    }
    __syncthreads();
    const float pr = prow[c];
    for (int r = 0; r < 64; ++r) {
      const int i = g * 64 + r;
      if (i != j) A[i * D_N + c] = fmaf(-colv[i], pr, A[i * D_N + c]);
    }
    __syncthreads();
    if (t < D_N) {
      A[j * D_N + t] = (t == j) ? dd : prow[t];    // new pivot row
      if (t != j) A[t * D_N + j] = -colv[t] * dd;  // new pivot column
    }
    __syncthreads();
  }
}

// ---------------------------------------------------------------------------
// Kernel 5: qm[c] = mu_c^T P mu_c.  grid = C_N, block = 256 (thread = d).
// ---------------------------------------------------------------------------
__global__ void k_qm(const float* __restrict__ P, const float* __restrict__ mu,
                     float* __restrict__ qm) {
  __shared__ float red[D_N];
  const int cIdx = blockIdx.x;
  const int d = threadIdx.x;
  const float* mc = mu + cIdx * D_N;
  float s = 0.0f;
  for (int e = 0; e < D_N; ++e) s = fmaf(P[d * D_N + e], mc[e], s);
  red[d] = s * mc[d];
  __syncthreads();
  for (int off = 128; off > 0; off >>= 1) {
    if (d < off) red[d] += red[d + off];
    __syncthreads();
  }
  if (d == 0) qm[cIdx] = red[0];
}

// ---------------------------------------------------------------------------
// Kernel 6: fused scores. Per workgroup: 16 rows of Z.
//   stage1: ZP = Ztile @ P   (16 waves, one 16-col block each, WMMA) -> LDS
//   stage2: cross = ZP @ muT (waves 0..3, WMMA) ; qz[r] per-row reduction
//   stage3: out[b,c] = logprior[c] - 0.5*(qz[b] - 2*cross + qm[c])
// grid = B/16 = 256 blocks x 512 threads (16 waves).
// ---------------------------------------------------------------------------
#define PITCH 260   // LDS row pitch in floats (bank-conflict free for stride-4 lanes)

__global__ void k_scores(const float* __restrict__ z, const float* __restrict__ P,
                         const float* __restrict__ muT, const float* __restrict__ qm,
                         const float* __restrict__ logprior, float* __restrict__ out) {
  __shared__ float Zs[16 * PITCH];
  __shared__ float ZPs[16 * PITCH];
  __shared__ float qzs[16];

  const int tid  = threadIdx.x;
  const int w    = tid >> 5;            // wave 0..15
  const int lane = tid & 31;
  const int m    = lane & 15;
  const int half = lane >> 4;
  const int b0   = blockIdx.x * 16;     // first batch row of this tile

  // Coalesced load of the 16x256 Z tile into LDS.
  for (int i = tid; i < 16 * D_N; i += 512) {
    const int r = i >> 8, cc = i & 255;
    Zs[r * PITCH + cc] = z[(b0 + r) * D_N + cc];
  }
  __syncthreads();

  // stage1: wave w computes ZP columns [16w, 16w+16), K = 256.
  {
    const int colBase = w * 16;
    v8f acc[2] = {};
    for (int k0 = 0; k0 < D_N; k0 += 8) {
#pragma unroll
      for (int u = 0; u < 2; ++u) {
        const int kk = k0 + u * 4 + half * 2;
        v2f a, b;
        a.x = Zs[m * PITCH + kk];
        a.y = Zs[m * PITCH + kk + 1];
        b.x = P[(kk    ) * D_N + colBase + m];
        b.y = P[(kk + 1) * D_N + colBase + m];
        acc[u] = wmma4(a, b, acc[u]);
      }
    }
    v8f t = acc[0] + acc[1];
#pragma unroll
    for (int v = 0; v < 8; ++v)
      ZPs[(v + 8 * half) * PITCH + colBase + m] = t[v];
  }
  __syncthreads();

  // qz: wave w reduces row w of (ZP .* Z).
  {
    float s = 0.0f;
    for (int cc = lane; cc < D_N; cc += 32)
      s = fmaf(ZPs[w * PITCH + cc], Zs[w * PITCH + cc], s);
    for (int off = 16; off > 0; off >>= 1) s += __shfl_xor(s, off, 32);
    if (lane == 0) qzs[w] = s;
  }

  // stage2: waves 0..3 compute cross tile (16 rows x 16 classes), K = 256.
  v8f cc2 = {};
  if (w < 4) {
    const int colBase = w * 16;         // class block
    v8f acc[2] = {};
    for (int k0 = 0; k0 < D_N; k0 += 8) {
#pragma unroll
      for (int u = 0; u < 2; ++u) {
        const int kk = k0 + u * 4 + half * 2;
        v2f a, b;
        a.x = ZPs[m * PITCH + kk];
        a.y = ZPs[m * PITCH + kk + 1];
        b.x = muT[(kk    ) * C_N + colBase + m];
        b.y = muT[(kk + 1) * C_N + colBase + m];
        acc[u] = wmma4(a, b, acc[u]);
      }
    }
    cc2 = acc[0] + acc[1];
  }
  __syncthreads();   // qzs visible

  // stage3: epilogue.
  if (w < 4) {
    const int cBase = w * 16;
    const int cIdx = cBase + m;
    const float lp = logprior[cIdx];
    const float qmc = qm[cIdx];
#pragma unroll
    for (int v = 0; v < 8; ++v) {
      const int row = v + 8 * half;
      const float m2 = qzs[row] - 2.0f * cc2[v] + qmc;
      out[(b0 + row) * C_N + cIdx] = lp - 0.5f * m2;
    }
  }
}

// ---------------------------------------------------------------------------
extern "C" void kernel_launch(void* const* d_in, const int* in_sizes, int n_in,
                              void* d_out, int out_size, void* d_ws, size_t ws_size,
                              hipStream_t stream) {
  const float* z = (const float*)d_in[0];   // (B, D) f32
  const int*   y = (const int*)d_in[1];     // (B,)   i32
  float* out = (float*)d_out;               // (B, C) f32

  float* ws = (float*)d_ws;
  float* mu       = ws;                  ws += C_N * D_N;   // 16384
  float* muT      = ws;                  ws += D_N * C_N;   // 16384
  float* wc       = ws;                  ws += C_N;         // 64
  float* logprior = ws;                  ws += C_N;         // 64
  float* G        = ws;                  ws += D_N * D_N;   // 65536
  float* covP     = ws;                  ws += D_N * D_N;   // 65536 (cov -> P in place)
  float* qm       = ws;                  ws += C_N;         // 64

  k_class_stats<<<C_N, 256, 0, stream>>>(z, y, mu, muT, wc, logprior);
  k_gram<<<32, 256, 0, stream>>>(z, G);                        // 256 waves = 256 tiles
  k_cov<<<D_N, 256, 0, stream>>>(G, mu, wc, covP);
  k_gj_inverse<<<1, 1024, 0, stream>>>(covP);                  // covP becomes precision P
  k_qm<<<C_N, 256, 0, stream>>>(covP, mu, qm);
  k_scores<<<B_N / 16, 512, 0, stream>>>(z, covP, muT, qm, logprior, out);
}